// GraphSAGE_86199993631208
// MI455X (gfx1250) — compile-verified
//
#include <hip/hip_runtime.h>

#define N_NODES 100000
#define N_EDGES 1600000
#define FDIM    128

typedef __attribute__((ext_vector_type(2))) float v2f;
typedef __attribute__((ext_vector_type(8))) float v8f;

// ---------------- degree kernels ----------------
__global__ void deg_count_kernel(const long long* __restrict__ dst,
                                 float* __restrict__ deg, int E) {
    int e = blockIdx.x * blockDim.x + threadIdx.x;
    if (e < E) atomicAdd(&deg[(int)dst[e]], 1.0f);
}

__global__ void invdeg_kernel(float* __restrict__ deg, int n) {
    int i = blockIdx.x * blockDim.x + threadIdx.x;
    if (i < n) deg[i] = 1.0f / fmaxf(deg[i], 1.0f);
}

// ---------------- COO scatter-add (mean numerator) ----------------
// grid.x covers edges, grid.y covers feature chunks of 16 floats.
// Indices loaded once per 16 features; atomics stay L2-resident (agg fits in 192MB L2).
__global__ void scatter_kernel(const float* __restrict__ x,
                               const long long* __restrict__ src,
                               const long long* __restrict__ dst,
                               float* __restrict__ agg, int E) {
    int e = blockIdx.x * blockDim.x + threadIdx.x;
    if (e >= E) return;
    int c = blockIdx.y << 4;
    int s = (int)src[e];
    int d = (int)dst[e];
    const float* xs = x + (size_t)s * FDIM + c;
    float*       a  = agg + (size_t)d * FDIM + c;
#pragma unroll
    for (int q = 0; q < 4; ++q) {
        const float4 v = *(const float4*)(xs + 4 * q);
        atomicAdd(a + 4 * q + 0, v.x);
        atomicAdd(a + 4 * q + 1, v.y);
        atomicAdd(a + 4 * q + 2, v.z);
        atomicAdd(a + 4 * q + 3, v.w);
    }
}

// ---------------- fused SAGE GEMM: out = (agg*invdeg) @ Wl^T + xin @ Wr^T + b ----------------
// One wave owns a 16-row x (NT*16)-col output stripe: A fragments (the expensive,
// N-scaled operand) are loaded once and reused across all NT column tiles; B
// (weights, 64KB, L0/L2-hot) is re-read per tile. 16 chained v_wmma per K-step.
template <int NT>
__global__ __launch_bounds__(256)
void sage_gemm_kernel(const float* __restrict__ agg,
                      const float* __restrict__ invdeg,
                      const float* __restrict__ xin,
                      const float* __restrict__ Wl,   // [Fout, K] row-major
                      const float* __restrict__ Wr,   // [Fout, K] row-major
                      const float* __restrict__ bias, // [Fout]
                      float* __restrict__ out,        // [N, Fout]
                      int N, int K, int do_relu) {
    constexpr int Fout = NT * 16;
    const int lane = threadIdx.x & 31;
    const int wave = threadIdx.x >> 5;
    const int mtile = blockIdx.x * 8 + wave;
    const int row0 = mtile * 16;
    if (row0 >= N) return;                 // uniform per wave: whole wave exits

    const int m    = lane & 15;            // A row / B col / D col owned by lane
    const int half = lane >> 4;            // K sub-offset selector

    const float inv = invdeg[row0 + m];
    const float* aRow = agg + (size_t)(row0 + m) * K + 2 * half;
    const float* xRow = xin + (size_t)(row0 + m) * K + 2 * half;
    const float* wlRow = Wl + (size_t)m * K + 2 * half;  // + nt*16*K per tile
    const float* wrRow = Wr + (size_t)m * K + 2 * half;

    v8f acc[NT];
#pragma unroll
    for (int nt = 0; nt < NT; ++nt) acc[nt] = {};

    for (int k0 = 0; k0 < K; k0 += 4) {
        v2f am = *(const v2f*)(aRow + k0);
        v2f ax = *(const v2f*)(xRow + k0);
        am *= inv;                         // mean = agg / max(deg,1)
#pragma unroll
        for (int nt = 0; nt < NT; ++nt) {
            v2f bl = *(const v2f*)(wlRow + (size_t)nt * 16 * K + k0);
            v2f br = *(const v2f*)(wrRow + (size_t)nt * 16 * K + k0);
            acc[nt] = __builtin_amdgcn_wmma_f32_16x16x4_f32(
                          false, am, false, bl, (short)0, acc[nt], false, false);
            acc[nt] = __builtin_amdgcn_wmma_f32_16x16x4_f32(
                          false, ax, false, br, (short)0, acc[nt], false, false);
        }
    }

#pragma unroll
    for (int nt = 0; nt < NT; ++nt) {
        const float bv = bias[nt * 16 + m];
        float* orow = out + (size_t)row0 * Fout + nt * 16 + m;
#pragma unroll
        for (int r = 0; r < 8; ++r) {
            float v = acc[nt][r] + bv;
            if (do_relu) v = fmaxf(v, 0.0f);
            orow[(size_t)(r + 8 * half) * Fout] = v;
        }
    }
}

// ---------------- launch ----------------
extern "C" void kernel_launch(void* const* d_in, const int* in_sizes, int n_in,
                              void* d_out, int out_size, void* d_ws, size_t ws_size,
                              hipStream_t stream) {
    const float*     x   = (const float*)d_in[0];
    const long long* ei  = (const long long*)d_in[1];   // int64 edge_index [2, E]
    const float*     Wl0 = (const float*)d_in[2];
    const float*     Wr0 = (const float*)d_in[3];
    const float*     b0  = (const float*)d_in[4];
    const float*     Wl1 = (const float*)d_in[5];
    const float*     Wr1 = (const float*)d_in[6];
    const float*     b1  = (const float*)d_in[7];
    const float*     Wl2 = (const float*)d_in[8];
    const float*     Wr2 = (const float*)d_in[9];
    const float*     b2  = (const float*)d_in[10];
    float*           out = (float*)d_out;

    const long long* src = ei;            // row 0
    const long long* dst = ei + N_EDGES;  // row 1

    // workspace layout
    char* ws = (char*)d_ws;
    size_t o = 0;
    float* invdeg = (float*)(ws + o); o += ((size_t)N_NODES * 4 + 255) & ~(size_t)255;
    float* agg    = (float*)(ws + o); o += (size_t)N_NODES * FDIM * 4;
    float* h1     = (float*)(ws + o); o += (size_t)N_NODES * FDIM * 4;
    float* h2     = (float*)(ws + o); o += (size_t)N_NODES * FDIM * 4;
    (void)ws_size; (void)in_sizes; (void)n_in; (void)out_size;

    const size_t aggBytes = (size_t)N_NODES * FDIM * 4;

    // degrees (shared by all layers)
    hipMemsetAsync(invdeg, 0, (size_t)N_NODES * 4, stream);
    deg_count_kernel<<<(N_EDGES + 255) / 256, 256, 0, stream>>>(dst, invdeg, N_EDGES);
    invdeg_kernel<<<(N_NODES + 255) / 256, 256, 0, stream>>>(invdeg, N_NODES);

    dim3 sgrid((N_EDGES + 255) / 256, FDIM / 16);
    const int mtiles = (N_NODES + 15) / 16;
    dim3 ggrid((mtiles + 7) / 8, 1);

    // layer 0: x -> h1 (relu)
    hipMemsetAsync(agg, 0, aggBytes, stream);
    scatter_kernel<<<sgrid, 256, 0, stream>>>(x, src, dst, agg, N_EDGES);
    sage_gemm_kernel<8><<<ggrid, 256, 0, stream>>>(agg, invdeg, x, Wl0, Wr0, b0,
                                                   h1, N_NODES, FDIM, 1);
    // layer 1: h1 -> h2 (relu)
    hipMemsetAsync(agg, 0, aggBytes, stream);
    scatter_kernel<<<sgrid, 256, 0, stream>>>(h1, src, dst, agg, N_EDGES);
    sage_gemm_kernel<8><<<ggrid, 256, 0, stream>>>(agg, invdeg, h1, Wl1, Wr1, b1,
                                                   h2, N_NODES, FDIM, 1);
    // layer 2: h2 -> out (no relu)
    hipMemsetAsync(agg, 0, aggBytes, stream);
    scatter_kernel<<<sgrid, 256, 0, stream>>>(h2, src, dst, agg, N_EDGES);
    sage_gemm_kernel<4><<<ggrid, 256, 0, stream>>>(agg, invdeg, h2, Wl2, Wr2, b2,
                                                   out, N_NODES, FDIM, 0);
}